// GatedGCNGraphGymLayer_50440095924338
// MI455X (gfx1250) — compile-verified
//
#include <hip/hip_runtime.h>

// GatedGCN layer for MI455X (gfx1250, wave32).
// fp32 WMMA (V_WMMA_F32_16X16X4_F32) for all 5 GEMMs. Weight-matrix B-fragments
// are hoisted into VGPRs (128 regs) outside the tile loops; K-loop is kc-outer /
// nt-inner with 4 live accumulators so dependent WMMAs are 4 apart (ISA 7.12.1).
// Fused edge pass with L2-resident gather/scatter atomics; LDS-reduced BN stats.

#define N_NODES 100000
#define N_EDGES 1000000
#define DIM 64
#define ND (N_NODES * DIM)                    // 6,400,000 floats
#define ED ((long long)N_EDGES * (long long)DIM)

typedef float v2f __attribute__((ext_vector_type(2)));
typedef float v8f __attribute__((ext_vector_type(8)));

// ---------------------------------------------------------------------------
// B-fragment layout for V_WMMA_F32_16X16X4_F32 (ISA 7.12.2): VGPR0 = K {0|2}
// row, VGPR1 = K {1|3} row, N striped over lanes 0-15 (mirrors A's K split
// across lane halves). bf[nt][kc] covers W rows kc*4..kc*4+3, cols nt*16+..15.
// ---------------------------------------------------------------------------
__device__ __forceinline__ void load_bfrags(const float* __restrict__ W, int lane,
                                            v2f (&bf)[4][16]) {
  const int l15 = lane & 15;
  const int hi  = lane >> 4;
  const float* bp0 = W + (hi * 2) * DIM + l15;
#pragma unroll
  for (int nt = 0; nt < 4; ++nt) {
#pragma unroll
    for (int kc = 0; kc < 16; ++kc) {
      const float* p = bp0 + nt * 16 + kc * 4 * DIM;
      bf[nt][kc].x = p[0];
      bf[nt][kc].y = p[DIM];
    }
  }
}

// All four 16x16 column blocks of (16 x 64) @ (64 x 64) in one pass.
// A-frag (ISA 7.12.2, 32-bit A 16x4): lanes 0-15 hold K=0,1; lanes 16-31 hold
// K=2,3 (one M-row per lane). C/D: VGPR r -> rows r / r+8, N = lane&15.
__device__ __forceinline__ void wmma_gemm_16x64_all(const float* __restrict__ Atile,
                                                    const v2f (&bf)[4][16], int lane,
                                                    v8f (&acc)[4]) {
  const int l15 = lane & 15;
  const int hi  = lane >> 4;
  const float* ap = Atile + l15 * DIM + hi * 2;
  const v8f vz = {};
#pragma unroll
  for (int nt = 0; nt < 4; ++nt) acc[nt] = vz;
#pragma unroll
  for (int kc = 0; kc < 16; ++kc) {
    v2f a;
    a.x = ap[kc * 4 + 0];
    a.y = ap[kc * 4 + 1];
#pragma unroll
    for (int nt = 0; nt < 4; ++nt) {
      acc[nt] = __builtin_amdgcn_wmma_f32_16x16x4_f32(false, a, false, bf[nt][kc],
                                                      (short)0, acc[nt], false, false);
    }
  }
}

// ---------------------------------------------------------------------------
__global__ void k_zero(float* __restrict__ p, long long n) {
  long long i = (long long)blockIdx.x * blockDim.x + threadIdx.x;
  if (i < n) p[i] = 0.0f;
}

// Bx, Dx, Ex = x @ {WB,WD,WE} + bias. One wave per (16-row tile, matrix).
__global__ void k_node_gemm3(const float* __restrict__ x,
                             const float* __restrict__ WB, const float* __restrict__ bB,
                             const float* __restrict__ WD, const float* __restrict__ bD,
                             const float* __restrict__ WE, const float* __restrict__ bE,
                             float* __restrict__ ws) {
  const int lane = threadIdx.x & 31;
  const int wave = threadIdx.x >> 5;
  const int nTiles = N_NODES / 16;                // 6250 (exact)
  int task = blockIdx.x * (blockDim.x >> 5) + wave;
  if (task >= nTiles * 3) return;
  const int which = task / nTiles;
  const int tile  = task - which * nTiles;
  const float* W; const float* b; float* out;
  if (which == 0)      { W = WB; b = bB; out = ws; }
  else if (which == 1) { W = WD; b = bD; out = ws + (long long)ND; }
  else                 { W = WE; b = bE; out = ws + 2LL * ND; }
  const int l15 = lane & 15, hi = lane >> 4;

  v2f bf[4][16];
  load_bfrags(W, lane, bf);
  v8f acc[4];
  wmma_gemm_16x64_all(x + (long long)tile * 16 * DIM, bf, lane, acc);
#pragma unroll
  for (int nt = 0; nt < 4; ++nt) {
    const int col = nt * 16 + l15;
    const float bias = b[col];
#pragma unroll
    for (int r = 0; r < 8; ++r) {
      const int row = tile * 16 + hi * 8 + r;
      out[(long long)row * DIM + col] = acc[nt][r] + bias;
    }
  }
}

// Edge pass: Ce = e@WC+bC (WMMA), e_ij = Dx[dst]+Ex[src]+Ce, sigma=sigmoid,
// scatter num/den atomics, store e_ij, accumulate BN stats for e.
__global__ void k_edge(const float* __restrict__ e, const int* __restrict__ ei,
                       const float* __restrict__ WC, const float* __restrict__ bC,
                       const float* __restrict__ Bx, const float* __restrict__ Dx,
                       const float* __restrict__ Ex,
                       float* __restrict__ num, float* __restrict__ den,
                       float* __restrict__ eout, float* __restrict__ stats) {
  __shared__ float s_red[128];
  if (threadIdx.x < 128) s_red[threadIdx.x] = 0.0f;
  __syncthreads();
  const int lane = threadIdx.x & 31;
  const int wave = threadIdx.x >> 5;
  const int wpb  = blockDim.x >> 5;
  const int l15 = lane & 15, hi = lane >> 4;
  const int nTiles = N_EDGES / 16;                // 62500 (exact)
  const int stride = gridDim.x * wpb;

  // Loop-invariant: weight fragments (128 VGPRs) and biases.
  v2f bf[4][16];
  load_bfrags(WC, lane, bf);
  float bias[4];
#pragma unroll
  for (int nt = 0; nt < 4; ++nt) bias[nt] = bC[nt * 16 + l15];

  float lsum[4] = {0.f, 0.f, 0.f, 0.f};
  float lssq[4] = {0.f, 0.f, 0.f, 0.f};
  for (int tile = blockIdx.x * wpb + wave; tile < nTiles; tile += stride) {
    if (tile + stride < nTiles)
      __builtin_prefetch(e + (long long)(tile + stride) * 16 * DIM, 0, 1);

    v8f acc[4];
    wmma_gemm_16x64_all(e + (long long)tile * 16 * DIM, bf, lane, acc);

    // Edge endpoints for this half-wave's 8 rows (uniform across 16 lanes).
    int srcN[8], dstN[8];
#pragma unroll
    for (int r = 0; r < 8; ++r) {
      const int edge = tile * 16 + hi * 8 + r;
      srcN[r] = ei[edge];                         // row 0: source j
      dstN[r] = ei[N_EDGES + edge];               // row 1: target i
    }
#pragma unroll
    for (int nt = 0; nt < 4; ++nt) {
      const int col = nt * 16 + l15;
#pragma unroll
      for (int r = 0; r < 8; ++r) {
        const int edge = tile * 16 + hi * 8 + r;
        const float eij = acc[nt][r] + bias[nt] +
                          Dx[(long long)dstN[r] * DIM + col] +
                          Ex[(long long)srcN[r] * DIM + col];
        const float sg = 1.0f / (1.0f + __expf(-eij));
        eout[(long long)edge * DIM + col] = eij;
        atomicAdd(&num[(long long)dstN[r] * DIM + col],
                  sg * Bx[(long long)srcN[r] * DIM + col]);
        atomicAdd(&den[(long long)dstN[r] * DIM + col], sg);
        lsum[nt] += eij;
        lssq[nt] += eij * eij;
      }
    }
  }
#pragma unroll
  for (int nt = 0; nt < 4; ++nt) {
    const int col = nt * 16 + l15;
    atomicAdd(&s_red[col], lsum[nt]);             // ds_add_f32
    atomicAdd(&s_red[64 + col], lssq[nt]);
  }
  __syncthreads();
  if (threadIdx.x < 128) atomicAdd(&stats[threadIdx.x], s_red[threadIdx.x]);
}

// Node pass: Ax = x@WA+bA (WMMA), h = Ax + num/(den+eps); store pre-BN h,
// accumulate BN stats for x.
__global__ void k_node_update(const float* __restrict__ x,
                              const float* __restrict__ WA, const float* __restrict__ bA,
                              const float* __restrict__ num, const float* __restrict__ den,
                              float* __restrict__ hout, float* __restrict__ statsX) {
  __shared__ float s_red[128];
  if (threadIdx.x < 128) s_red[threadIdx.x] = 0.0f;
  __syncthreads();
  const int lane = threadIdx.x & 31;
  const int wave = threadIdx.x >> 5;
  const int wpb  = blockDim.x >> 5;
  const int l15 = lane & 15, hi = lane >> 4;
  const int nTiles = N_NODES / 16;                // 6250 (exact)
  const int stride = gridDim.x * wpb;

  v2f bf[4][16];
  load_bfrags(WA, lane, bf);
  float bias[4];
#pragma unroll
  for (int nt = 0; nt < 4; ++nt) bias[nt] = bA[nt * 16 + l15];

  float lsum[4] = {0.f, 0.f, 0.f, 0.f};
  float lssq[4] = {0.f, 0.f, 0.f, 0.f};
  for (int tile = blockIdx.x * wpb + wave; tile < nTiles; tile += stride) {
    v8f acc[4];
    wmma_gemm_16x64_all(x + (long long)tile * 16 * DIM, bf, lane, acc);
#pragma unroll
    for (int nt = 0; nt < 4; ++nt) {
      const int col = nt * 16 + l15;
#pragma unroll
      for (int r = 0; r < 8; ++r) {
        const long long idx = (long long)(tile * 16 + hi * 8 + r) * DIM + col;
        const float h = acc[nt][r] + bias[nt] + num[idx] / (den[idx] + 1e-6f);
        hout[idx] = h;
        lsum[nt] += h;
        lssq[nt] += h * h;
      }
    }
  }
#pragma unroll
  for (int nt = 0; nt < 4; ++nt) {
    const int col = nt * 16 + l15;
    atomicAdd(&s_red[col], lsum[nt]);
    atomicAdd(&s_red[64 + col], lssq[nt]);
  }
  __syncthreads();
  if (threadIdx.x < 128) atomicAdd(&statsX[threadIdx.x], s_red[threadIdx.x]);
}

// Fold BN stats into per-channel scale/shift: out = v*scale[c] + shift[c].
__global__ void k_finalize(const float* __restrict__ stats,
                           const float* __restrict__ gx, const float* __restrict__ bx,
                           const float* __restrict__ ge, const float* __restrict__ be,
                           float* __restrict__ coef) {
  const int c = threadIdx.x;
  if (c >= DIM) return;
  const float invE = 1.0f / (float)N_EDGES;
  const float mue  = stats[c] * invE;
  const float vare = stats[64 + c] * invE - mue * mue;
  const float se   = ge[c] * rsqrtf(vare + 1e-5f);
  coef[c]      = se;
  coef[64 + c] = be[c] - mue * se;
  const float invN = 1.0f / (float)N_NODES;
  const float mux  = stats[128 + c] * invN;
  const float varx = stats[192 + c] * invN - mux * mux;
  const float sx   = gx[c] * rsqrtf(varx + 1e-5f);
  coef[128 + c] = sx;
  coef[192 + c] = bx[c] - mux * sx;
}

__global__ void k_norm_relu(float* __restrict__ buf, const float* __restrict__ coef,
                            long long n) {
  long long i = (long long)blockIdx.x * blockDim.x + threadIdx.x;
  const long long stride = (long long)gridDim.x * blockDim.x;
  for (; i < n; i += stride) {
    const int c = (int)(i & (long long)(DIM - 1));
    const float v = buf[i] * coef[c] + coef[64 + c];
    buf[i] = fmaxf(v, 0.0f);
  }
}

// ---------------------------------------------------------------------------
extern "C" void kernel_launch(void* const* d_in, const int* in_sizes, int n_in,
                              void* d_out, int out_size, void* d_ws, size_t ws_size,
                              hipStream_t stream) {
  const float* x  = (const float*)d_in[0];
  const float* e  = (const float*)d_in[1];
  const int*   ei = (const int*)d_in[2];
  const float* WA = (const float*)d_in[3];  const float* bA = (const float*)d_in[4];
  const float* WB = (const float*)d_in[5];  const float* bB = (const float*)d_in[6];
  const float* WC = (const float*)d_in[7];  const float* bC = (const float*)d_in[8];
  const float* WD = (const float*)d_in[9];  const float* bD = (const float*)d_in[10];
  const float* WE = (const float*)d_in[11]; const float* bE = (const float*)d_in[12];
  const float* gx = (const float*)d_in[13]; const float* bx = (const float*)d_in[14];
  const float* ge = (const float*)d_in[15]; const float* be = (const float*)d_in[16];

  // Workspace layout (floats): Bx | Dx | Ex | num | den | stats(256) | coef(256)
  float* ws    = (float*)d_ws;
  float* Bx    = ws;
  float* Dx    = ws + (long long)ND;
  float* Ex    = ws + 2LL * ND;
  float* num   = ws + 3LL * ND;
  float* den   = ws + 4LL * ND;
  float* stats = ws + 5LL * ND;
  float* coef  = stats + 256;

  float* xout = (float*)d_out;                  // N*D floats
  float* eout = (float*)d_out + (long long)ND;  // E*D floats

  // 1) zero num/den/stats (every launch: deterministic, graph-safe)
  {
    const long long nz = 2LL * ND + 256;
    k_zero<<<(int)((nz + 255) / 256), 256, 0, stream>>>(num, nz);
  }
  // 2) node GEMMs: Bx, Dx, Ex
  {
    const int tasks = (N_NODES / 16) * 3;       // 18750 wave-tasks
    const int wpb = 8;                          // 8 waves / 256-thread block
    k_node_gemm3<<<(tasks + wpb - 1) / wpb, 32 * wpb, 0, stream>>>(
        x, WB, bB, WD, bD, WE, bE, ws);
  }
  // 3) fused edge pass (Ce GEMM + gather + gate + scatter + e-stats)
  k_edge<<<1024, 256, 0, stream>>>(e, ei, WC, bC, Bx, Dx, Ex, num, den, eout, stats);
  // 4) fused node pass (Ax GEMM + gated mean + x-stats)
  k_node_update<<<256, 256, 0, stream>>>(x, WA, bA, num, den, xout, stats + 128);
  // 5) BN coefficient finalize
  k_finalize<<<1, 64, 0, stream>>>(stats, gx, bx, ge, be, coef);
  // 6) in-place BN+ReLU
  k_norm_relu<<<4096, 256, 0, stream>>>(eout, coef, ED);
  k_norm_relu<<<512, 256, 0, stream>>>(xout, coef + 128, (long long)ND);
}